// inversedtcwtLayer_70437463654713
// MI455X (gfx1250) — compile-verified
//
#include <hip/hip_runtime.h>
#include <hip/hip_bf16.h>

// ---------------------------------------------------------------------------
// Inverse DTCWT for MI455X (gfx1250, wave32).
//
// Pass 1 (k_level1, VALU): qshift level-1 synthesis fused (col+row colifilt,
//   c2q assembly of high1 subbands) -> Z (8,512,512,3) staged in d_ws.
//   Gather-formulated 2x upsampling with dtcwt reflect indexing; LDS holds the
//   per-tile y1/y2 row-window so each wave does the row-direction colifilt
//   from shared data.
//
// Pass 2 (k_level0, WMMA): biort level-0 synthesis as banded 16x16 GEMMs with
//   V_WMMA_F32_16X16X4_F32. Per 16x16 output tile:
//     column pass: Y1 = H0*Zp + H1*LHp ; Y2 = H0*HLp + H1*HHp  (K=24 -> 6 wmma
//                  steps, 2 N-subtiles to produce the 32-col halo for rows)
//     row pass:    Out = Y1p*G0 + Y2p*G1                       (K=24)
//   The banded filter fragment g[k - m - 1] is the same per-lane value whether
//   used as A (col pass) or B (row pass), so 12 constant v2f fragments are
//   shared. Z stays resident in the 192MB L2 between the two kernels.
//
// WMMA layout assumptions (ISA 7.12.2, f32 16x16x4):
//   A 16x4 : lanes 0-15 rows M=0..15 hold {K0,K1}; lanes 16-31 hold {K2,K3}
//   B 4x16 : lanes 0-15 cols N=0..15 hold {K0,K1}; lanes 16-31 hold {K2,K3}
//   C/D    : vgpr v, lanes 0-15 -> (M=v, N=lane); lanes 16-31 -> (M=v+8)
// ---------------------------------------------------------------------------

typedef float v2f __attribute__((ext_vector_type(2)));
typedef float v8f __attribute__((ext_vector_type(8)));

#define SCQ 0.70710678118654752440f

// ---- qshift synthesis filters (14-tap) ----
__constant__ float cG0B[14] = {  // = H0A
   0.00325314f, -0.00388321f,  0.03466035f, -0.03887280f,
  -0.11720389f,  0.27529538f,  0.75614564f,  0.56881042f,
   0.01186609f, -0.10671180f,  0.02382538f,  0.01702522f,
  -0.00543948f, -0.00455690f };
__constant__ float cG0A[14] = {  // = reverse(H0A)
  -0.00455690f, -0.00543948f,  0.01702522f,  0.02382538f,
  -0.10671180f,  0.01186609f,  0.56881042f,  0.75614564f,
   0.27529538f, -0.11720389f, -0.03887280f,  0.03466035f,
  -0.00388321f,  0.00325314f };
__constant__ float cG1B[14] = {  // = H1A = reverse(H0A) * (-1)^i
  -0.00455690f,  0.00543948f,  0.01702522f, -0.02382538f,
  -0.10671180f, -0.01186609f,  0.56881042f, -0.75614564f,
   0.27529538f,  0.11720389f, -0.03887280f, -0.03466035f,
  -0.00388321f, -0.00325314f };
__constant__ float cG1A[14] = {  // = reverse(H1A)
  -0.00325314f, -0.00388321f, -0.03466035f, -0.03887280f,
   0.11720389f,  0.27529538f, -0.75614564f,  0.56881042f,
  -0.01186609f, -0.10671180f, -0.02382538f,  0.01702522f,
   0.00543948f, -0.00455690f };

// ---- biort synthesis filters (symmetric -> conv == corr) ----
__constant__ float cG0O[7] = { -3.0f/280.0f, -15.0f/280.0f, 73.0f/280.0f,
                               170.0f/280.0f, 73.0f/280.0f, -15.0f/280.0f,
                               -3.0f/280.0f };
__constant__ float cG1O[5] = { -1.0f/20.0f, -5.0f/20.0f, 12.0f/20.0f,
                               -5.0f/20.0f, -1.0f/20.0f };

__device__ __forceinline__ int ref_idx(int i, int L) {
  if (i < 0)  i = -1 - i;          // symmetric extension, single reflection
  if (i >= L) i = 2 * L - 1 - i;   // (halo <= 8 << L)
  return i;
}

// c2q gather: value of the double-size quad image at (r, v) for subband `sub`
// (0=lh pair(0,5), 1=hl pair(2,3), 2=hh pair(1,4)) of a (.., Ld, Ld, 36) tensor.
__device__ __forceinline__ float c2q_fetch(const float* __restrict__ hi,
                                           int b, int r, int v, int ch,
                                           int sub, int Ld) {
  constexpr int SA[3] = { 0, 2, 1 };
  constexpr int SB[3] = { 5, 3, 4 };
  const int i = r >> 1, j = v >> 1;
  const long long base = (((long long)b * Ld + i) * Ld + j) * 36;
  const int py = r & 1, px = v & 1;
  const int im = (px ^ py) ? 18 : 0;                  // (0,1)/(1,0) use imag
  const float a  = hi[base + ch + 3 * SA[sub] + im];
  const float bb = hi[base + ch + 3 * SB[sub] + im];
  // py==0: (a+b)*SC ; py==1,px==0: (a-b)*SC ; py==1,px==1: (b-a)*SC
  return py ? ((px ^ py) ? (a - bb) * SCQ : (bb - a) * SCQ) : (a + bb) * SCQ;
}

// One output sample of _colifilt along rows at output row y=4*qy+sy, column v.
// g1=false -> (G0B,G0A) pair (sum>0: ha base 6, hb base 7)
// g1=true  -> (G1B,G1A) pair (sum<0: ha base 7, hb base 6)
// sub = -1 -> buf is `low` (256x256x3); else c2q of high1 (128-grid).
__device__ __forceinline__ float colifilt_eval(bool g1, int qy, int sy, int v,
                                               const float* __restrict__ buf,
                                               int b, int ch, int sub) {
  const float* ha = g1 ? cG1B : cG0B;
  const float* hb = g1 ? cG1A : cG0A;
  const int   p   = sy >> 1;
  const bool  aB  = (sy & 1) == 0;
  const float* f  = aB ? ha : hb;
  const int base  = g1 ? (aB ? 7 : 6) : (aB ? 6 : 7);
  float acc = 0.0f;
#pragma unroll
  for (int j = 0; j < 7; ++j) {
    const int r = ref_idx(2 * qy + base - 2 * j, 256);
    float x;
    if (sub < 0) x = buf[(((long long)b * 256 + r) * 256 + v) * 3 + ch];
    else         x = c2q_fetch(buf, b, r, v, ch, sub, 128);
    acc += f[2 * j + p] * x;
  }
  return acc;
}

// ============================ level-1 (qshift) =============================
// Z[b,y,x,ch] for one 16x16 tile per block; 256 threads.
__global__ void __launch_bounds__(256)
k_level1(const float* __restrict__ low, const float* __restrict__ high1,
         float* __restrict__ Z) {
  const int tx = blockIdx.x, ty = blockIdx.y, bz = blockIdx.z;
  const int b = bz / 3, ch = bz % 3;
  const int x0 = tx * 16, y0 = ty * 16;
  const int u0 = 8 * tx - 6;            // column window in the 256-col domain

  __shared__ float y1s[16][21];
  __shared__ float y2s[16][21];

  const int tid = threadIdx.x;
  // Stage A: y1/y2 at 16 rows x 20 (reflected) columns.
  for (int idx = tid; idx < 2 * 16 * 20; idx += 256) {
    const int plane = idx / 320;
    const int rem   = idx - plane * 320;
    const int ry = rem / 20, ci = rem % 20;
    const int v  = ref_idx(u0 + ci, 256);
    const int y  = y0 + ry, qy = y >> 2, sy = y & 3;
    float val;
    if (plane == 0)
      val = colifilt_eval(false, qy, sy, v, low,   b, ch, -1) +
            colifilt_eval(true,  qy, sy, v, high1, b, ch,  0);
    else
      val = colifilt_eval(false, qy, sy, v, high1, b, ch,  1) +
            colifilt_eval(true,  qy, sy, v, high1, b, ch,  2);
    (plane ? y2s : y1s)[ry][ci] = val;
  }
  __syncthreads();

  // Stage B: row-direction colifilt from LDS (no reflection needed: stage A
  // already evaluated at reflected columns).
  const int ry = tid >> 4, cx = tid & 15;
  const int x  = x0 + cx;
  const int q2 = 2 * (cx >> 2);
  const int sx = x & 3, p = sx >> 1;
  const bool aB = (sx & 1) == 0;
  const float* f1 = aB ? cG0B : cG0A;  const int b1 = aB ? 6 : 7;  // y1: G0 pair
  const float* f2 = aB ? cG1B : cG1A;  const int b2 = aB ? 7 : 6;  // y2: G1 pair
  float acc = 0.0f;
#pragma unroll
  for (int j = 0; j < 7; ++j) {
    acc += f1[2 * j + p] * y1s[ry][q2 + b1 + 6 - 2 * j];
    acc += f2[2 * j + p] * y2s[ry][q2 + b2 + 6 - 2 * j];
  }
  Z[(((long long)b * 512 + (y0 + ry)) * 512 + x) * 3 + ch] = acc;
}

// ============================ level-0 (biort, WMMA) ========================
__device__ __forceinline__ float tap7(int i) {
  int ic = i < 0 ? 0 : (i > 6 ? 6 : i);
  float t = cG0O[ic];
  return (i == ic) ? t : 0.0f;
}
__device__ __forceinline__ float tap5(int i) {
  int ic = i < 0 ? 0 : (i > 4 ? 4 : i);
  float t = cG1O[ic];
  return (i == ic) ? t : 0.0f;
}

__device__ __forceinline__ v8f wmma4(v2f a, v2f b, v8f c) {
  // D = A(16x4) * B(4x16) + C ; f32 throughout.
  return __builtin_amdgcn_wmma_f32_16x16x4_f32(
      /*neg_a=*/false, a, /*neg_b=*/false, b,
      /*c_mod=*/(short)0, c, /*reuse_a=*/false, /*reuse_b=*/false);
}

// One 16x16 output tile per 32-thread (1-wave) block.
__global__ void __launch_bounds__(32)
k_level0(const float* __restrict__ Z, const float* __restrict__ high0,
         float* __restrict__ out) {
  const int tx = blockIdx.x, ty = blockIdx.y, bz = blockIdx.z;
  const int b = bz / 3, ch = bz % 3;
  const int x0 = tx * 16, y0 = ty * 16;

  __shared__ float zs [24][33];
  __shared__ float lhs[24][33];
  __shared__ float hls[24][33];
  __shared__ float hhs[24][33];
  __shared__ float y1s[16][33];
  __shared__ float y2s[16][33];

  const int lane = threadIdx.x;
  const int lo = lane & 15;
  const int h2 = (lane >> 4) * 2;       // K sub-offset for this half-wave

  // Constant banded-filter fragments: value g[k - idx - 1] (7-tap) and
  // g[k - idx - 2] (5-tap) where idx = lane's M (as A) or N (as B) -- same
  // per-lane value in both roles, so shared between column and row passes.
  v2f f0[6], f1[6];
#pragma unroll
  for (int kk = 0; kk < 6; ++kk) {
    const int k0 = 4 * kk + h2;
    f0[kk].x = tap7(k0 - lo - 1);  f0[kk].y = tap7(k0 - lo);
    f1[kk].x = tap5(k0 - lo - 2);  f1[kk].y = tap5(k0 - lo - 1);
  }

  // Load padded tiles: rows y0-4..y0+19 (24), cols x0-8..x0+23 (32).
  for (int idx = lane; idx < 24 * 32; idx += 32) {
    const int r = idx >> 5, c = idx & 31;
    const int rr = ref_idx(y0 - 4 + r, 512);
    const int cc = ref_idx(x0 - 8 + c, 512);
    zs [r][c] = Z[(((long long)b * 512 + rr) * 512 + cc) * 3 + ch];
    lhs[r][c] = c2q_fetch(high0, b, rr, cc, ch, 0, 256);
    hls[r][c] = c2q_fetch(high0, b, rr, cc, ch, 1, 256);
    hhs[r][c] = c2q_fetch(high0, b, rr, cc, ch, 2, 256);
  }
  __syncthreads();

  // Column pass: Y1 = H0*Zp + H1*LHp ; Y2 = H0*HLp + H1*HHp, two 16-col
  // subtiles to cover the 32-col halo needed by the row pass.
#pragma unroll
  for (int t = 0; t < 2; ++t) {
    v8f a1 = {}; v8f a2 = {};
#pragma unroll
    for (int kk = 0; kk < 6; ++kk) {
      const int k0 = 4 * kk + h2;
      const int cc = 16 * t + lo;
      v2f bz = { zs [k0][cc], zs [k0 + 1][cc] };
      v2f bl = { lhs[k0][cc], lhs[k0 + 1][cc] };
      v2f bh = { hls[k0][cc], hls[k0 + 1][cc] };
      v2f bq = { hhs[k0][cc], hhs[k0 + 1][cc] };
      a1 = wmma4(f0[kk], bz, a1);
      a1 = wmma4(f1[kk], bl, a1);
      a2 = wmma4(f0[kk], bh, a2);
      a2 = wmma4(f1[kk], bq, a2);
    }
#pragma unroll
    for (int v = 0; v < 8; ++v) {
      const int row = v + (h2 << 2);        // +8 for lanes 16-31
      y1s[row][16 * t + lo] = a1[v];
      y2s[row][16 * t + lo] = a2[v];
    }
  }
  __syncthreads();

  // Row pass: Out = Y1p * G0mat + Y2p * G1mat (K window = stored cols 4..27).
  v8f o = {};
#pragma unroll
  for (int kk = 0; kk < 6; ++kk) {
    const int k0 = 4 * kk + h2;
    v2f A1 = { y1s[lo][k0 + 4], y1s[lo][k0 + 5] };
    v2f A2 = { y2s[lo][k0 + 4], y2s[lo][k0 + 5] };
    o = wmma4(A1, f0[kk], o);
    o = wmma4(A2, f1[kk], o);
  }
#pragma unroll
  for (int v = 0; v < 8; ++v) {
    const int row = y0 + v + (h2 << 2);
    const int col = x0 + lo;
    out[(((long long)b * 512 + row) * 512 + col) * 3 + ch] = o[v];
  }
}

// ===========================================================================
extern "C" void kernel_launch(void* const* d_in, const int* in_sizes, int n_in,
                              void* d_out, int out_size, void* d_ws, size_t ws_size,
                              hipStream_t stream) {
  (void)in_sizes; (void)n_in; (void)out_size; (void)ws_size;
  const float* low   = (const float*)d_in[0];   // (8,256,256,3)
  const float* high0 = (const float*)d_in[1];   // (8,256,256,36)
  const float* high1 = (const float*)d_in[2];   // (8,128,128,36)
  float* out = (float*)d_out;                   // (8,512,512,3)
  float* Zws = (float*)d_ws;                    // 24 MB intermediate, L2-resident

  dim3 grid(32, 32, 8 * 3);                     // 16x16 tiles, per (b, channel)
  k_level1<<<grid, dim3(256), 0, stream>>>(low, high1, Zws);
  k_level0<<<grid, dim3(32), 0, stream>>>(Zws, high0, out);
}